// EdgeOrient_22093311771174
// MI455X (gfx1250) — compile-verified
//
#include <hip/hip_runtime.h>
#include <hip/hip_bf16.h>

typedef __attribute__((ext_vector_type(2))) float v2f;
typedef __attribute__((ext_vector_type(8))) float v8f;

#define FDIM 128

// ---------------------------------------------------------------------------
// zero fill
// ---------------------------------------------------------------------------
__global__ void zero_kernel(float* __restrict__ p, size_t nfloats) {
    size_t i = ((size_t)blockIdx.x * blockDim.x + threadIdx.x) * 4;
    if (i + 3 < nfloats) {
        *(float4*)(p + i) = make_float4(0.f, 0.f, 0.f, 0.f);
    } else {
        for (; i < nfloats; ++i) p[i] = 0.f;
    }
}

// ---------------------------------------------------------------------------
// edge scatter: out[dst] += x[src] * orient   (one wave32 == one edge row)
// gid = e*32 + c ; c indexes a float4 chunk of the 128-float feature row
// ---------------------------------------------------------------------------
__global__ void scatter_kernel(float* __restrict__ out,
                               const float* __restrict__ x,
                               const int* __restrict__ idx,   // [2][E]
                               const float* __restrict__ orient,
                               long long E) {
    long long gid = (long long)blockIdx.x * blockDim.x + threadIdx.x;
    long long e = gid >> 5;
    if (e >= E) return;
    int c = (int)(gid & 31);
    long long src = (long long)idx[e];
    long long dst = (long long)idx[E + e];
    float o = orient[e];
    const float4 v = *(const float4*)(x + src * FDIM + c * 4);
    float* p = out + dst * FDIM + c * 4;
    unsafeAtomicAdd(p + 0, v.x * o);
    unsafeAtomicAdd(p + 1, v.y * o);
    unsafeAtomicAdd(p + 2, v.z * o);
    unsafeAtomicAdd(p + 3, v.w * o);
}

// ---------------------------------------------------------------------------
// out[m,n] = Aup[m,:]@Wup + Adn[m,:]@Wdn + Aself[m,:]@Wself
// Full f32 via V_WMMA_F32_16X16X4_F32.  Weights (3 x 64KB) live in LDS.
// Block = 256 threads = 8 waves; wave w computes column tile [16w,16w+16).
// A-fragment layout (16x4 f32): lanes 0-15 -> row=lane, K pair {k0,k0+1};
// lanes 16-31 -> row=lane-16, K pair {k0+2,k0+3}. B mirrors with N across lanes.
// ---------------------------------------------------------------------------
__global__ __launch_bounds__(256) void gemm3_wmma(
    float* __restrict__ out,
    const float* __restrict__ Aup,
    const float* __restrict__ Adn,
    const float* __restrict__ Aself,
    const float* __restrict__ Wup,
    const float* __restrict__ Wdn,
    const float* __restrict__ Wself,
    int nrows) {
    extern __shared__ float lds[];        // 3 * 128 * 128 floats = 192 KB
    float* w0 = lds;
    float* w1 = lds + FDIM * FDIM;
    float* w2 = lds + 2 * FDIM * FDIM;
    for (int i = threadIdx.x; i < FDIM * FDIM; i += blockDim.x) {
        w0[i] = Wup[i];
        w1[i] = Wdn[i];
        w2[i] = Wself[i];
    }
    __syncthreads();

    const int wave = threadIdx.x >> 5;
    const int lane = threadIdx.x & 31;
    const int lo = lane & 15;          // row (A) / col (B,C)
    const int hi = lane >> 4;          // K-pair selector
    const int n0 = wave * 16;

    const float* As[3] = {Aup, Adn, Aself};
    const float* Ws[3] = {w0, w1, w2};

    const int ntiles = nrows >> 4;     // nrows is a multiple of 16
    for (int t = blockIdx.x; t < ntiles; t += gridDim.x) {
        const int m0 = t << 4;

        // prefetch next tile's A rows into cache
        if (t + gridDim.x < ntiles) {
            const size_t mn = (size_t)((t + gridDim.x) << 4) + lo;
            __builtin_prefetch(Aup + mn * FDIM, 0, 1);
            __builtin_prefetch(Adn + mn * FDIM, 0, 1);
            __builtin_prefetch(Aself + mn * FDIM, 0, 1);
        }

        v8f acc = {};
#pragma unroll
        for (int s = 0; s < 3; ++s) {
            const float* __restrict__ arow = As[s] + (size_t)(m0 + lo) * FDIM;
            const float* __restrict__ W = Ws[s];
            for (int k0 = 0; k0 < FDIM; k0 += 4) {
                const int ka = k0 + 2 * hi;               // even -> 8B aligned
                v2f a = *(const v2f*)(arow + ka);          // global_load_b64
                v2f b;
                b.x = W[ka * FDIM + n0 + lo];              // ds_load_b32
                b.y = W[(ka + 1) * FDIM + n0 + lo];        // ds_load_b32
                acc = __builtin_amdgcn_wmma_f32_16x16x4_f32(
                    false, a, false, b, (short)0, acc, false, false);
            }
        }

        // C/D layout: lane -> col n0+lo ; vgpr r -> row r + 8*hi
        float* __restrict__ orow = out + (size_t)m0 * FDIM + n0 + lo;
#pragma unroll
        for (int r = 0; r < 8; ++r) {
            orow[(size_t)(r + 8 * hi) * FDIM] = acc[r];
        }
    }
}

// ---------------------------------------------------------------------------
// pooled[batch[i]] += |x[i]|   (one wave32 == one node row)
// ---------------------------------------------------------------------------
__global__ void pool_kernel(float* __restrict__ pooled,
                            const float* __restrict__ x,
                            const int* __restrict__ batch, int n) {
    long long gid = (long long)blockIdx.x * blockDim.x + threadIdx.x;
    long long i = gid >> 5;
    if (i >= n) return;
    int c = (int)(gid & 31);
    int b = batch[i];
    const float4 v = *(const float4*)(x + i * FDIM + c * 4);
    float* p = pooled + (size_t)b * FDIM + c * 4;
    unsafeAtomicAdd(p + 0, fabsf(v.x));
    unsafeAtomicAdd(p + 1, fabsf(v.y));
    unsafeAtomicAdd(p + 2, fabsf(v.z));
    unsafeAtomicAdd(p + 3, fabsf(v.w));
}

// ---------------------------------------------------------------------------
// tiny MLP head
// ---------------------------------------------------------------------------
__global__ void mlp1_kernel(float* __restrict__ h, const float* __restrict__ pooled,
                            const float* __restrict__ W1, const float* __restrict__ b1) {
    int row = blockIdx.x;      // 64
    int col = threadIdx.x;     // 128
    float s = b1[col];
    for (int k = 0; k < FDIM; ++k) s += pooled[row * FDIM + k] * W1[k * FDIM + col];
    h[row * FDIM + col] = fmaxf(s, 0.f);
}

__global__ void mlp2_kernel(float* __restrict__ out, const float* __restrict__ h,
                            const float* __restrict__ W2, const float* __restrict__ b2) {
    int gid = blockIdx.x * blockDim.x + threadIdx.x;
    if (gid >= 64 * 10) return;
    int row = gid / 10, col = gid % 10;
    float s = b2[col];
    for (int k = 0; k < FDIM; ++k) s += h[row * FDIM + k] * W2[k * 10 + col];
    out[gid] = s;
}

// ---------------------------------------------------------------------------
extern "C" void kernel_launch(void* const* d_in, const int* in_sizes, int n_in,
                              void* d_out, int out_size, void* d_ws, size_t ws_size,
                              hipStream_t stream) {
    const float* x     = (const float*)d_in[0];
    const int* up_idx  = (const int*)d_in[1];
    const int* dn_idx  = (const int*)d_in[2];
    const float* uo    = (const float*)d_in[3];
    const float* dno   = (const float*)d_in[4];
    const int* batch   = (const int*)d_in[5];
    const float* Wup   = (const float*)d_in[6];
    const float* Wdn   = (const float*)d_in[7];
    const float* Wself = (const float*)d_in[8];
    const float* W1    = (const float*)d_in[9];
    const float* b1    = (const float*)d_in[10];
    const float* W2    = (const float*)d_in[11];
    const float* b2    = (const float*)d_in[12];

    const int n = in_sizes[0] / FDIM;          // 100000
    const long long E = in_sizes[1] / 2;       // 1600000

    const size_t nf = (size_t)n * FDIM;
    float* ws     = (float*)d_ws;
    float* xA     = ws;                        // n*128
    float* xB     = xA + nf;                   // n*128
    float* up     = xB + nf;                   // n*128
    float* dn     = up + nf;                   // n*128 (contiguous after up)
    float* pooled = dn + nf;                   // 64*128
    float* h      = pooled + 64 * FDIM;        // 64*128

    hipMemcpyAsync(xA, x, nf * sizeof(float), hipMemcpyDeviceToDevice, stream);

    const long long sc_threads = E * 32;
    const int sc_blocks = (int)((sc_threads + 255) / 256);
    const size_t zcount = 2 * nf;
    const int z_blocks = (int)((zcount / 4 + 255) / 256);

    float* cur = xA;
    float* nxt = xB;
    for (int l = 0; l < 3; ++l) {
        zero_kernel<<<z_blocks, 256, 0, stream>>>(up, zcount);  // zeros up+dn
        scatter_kernel<<<sc_blocks, 256, 0, stream>>>(up, cur, up_idx, uo, E);
        scatter_kernel<<<sc_blocks, 256, 0, stream>>>(dn, cur, dn_idx, dno, E);
        const size_t off = (size_t)l * FDIM * FDIM;
        gemm3_wmma<<<512, 256, 3 * FDIM * FDIM * sizeof(float), stream>>>(
            nxt, up, dn, cur, Wup + off, Wdn + off, Wself + off, n);
        float* t = cur; cur = nxt; nxt = t;
    }

    zero_kernel<<<(64 * FDIM / 4 + 255) / 256, 256, 0, stream>>>(pooled, 64 * FDIM);
    const long long pl_threads = (long long)n * 32;
    pool_kernel<<<(int)((pl_threads + 255) / 256), 256, 0, stream>>>(pooled, cur, batch, n);
    mlp1_kernel<<<64, 128, 0, stream>>>(h, pooled, W1, b1);
    mlp2_kernel<<<3, 256, 0, stream>>>((float*)d_out, h, W2, b2);
}